// E3TAOBackflow_42374147342465
// MI455X (gfx1250) — compile-verified
//
#include <hip/hip_runtime.h>
#include <math.h>

// ---------------------------------------------------------------------------
// E3TAO backflow, fused single kernel for MI455X (gfx1250).
//
// Antisymmetry shortcut (exact, bitwise): the scalar branch of anti_mlp
// cancels to 0 and the vector branch equals one forward pass:
//   Vh0 = Wh0^T v ; vn = ||Vh0||_t ; gate = sigmoid([s;vn] @ Wm0 + bm0)
//   v'  = (Wu0^T Vh0) * gate ; Vh1 = Wh1^T v' ; Vu1 = Wu1^T Vh1
//   vo  = W1^T Vu1 * (1/sqrt(128))
// => Wm1, bm1, W0 are dead; scalar output slots are exact zeros.
//
// GEMMs run on V_WMMA_F32_16X16X4_F32 (full fp32 = reference math).
// Weight staging runs on the Tensor Data Mover (TENSOR_LOAD_TO_LDS),
// double-buffered in two 64 KB LDS buffers so the DMA of stage i+1 weights
// overlaps stage i compute; sync via s_wait_tensorcnt + workgroup barrier.
// ---------------------------------------------------------------------------

#define S_DIM  128
#define TOK    16              // tokens per workgroup
#define NCV    (TOK * 3)       // 48 activation columns (token, xyz)
#define MOUT   256
#define THREADS 256

typedef float v2f __attribute__((ext_vector_type(2)));
typedef float v8f __attribute__((ext_vector_type(8)));
typedef unsigned int u32x4 __attribute__((ext_vector_type(4)));
typedef int i32x4 __attribute__((ext_vector_type(4)));
typedef int i32x8 __attribute__((ext_vector_type(8)));

__device__ __forceinline__ v8f wmma_f32_k4(v2f a, v2f b, v8f c) {
  return __builtin_amdgcn_wmma_f32_16x16x4_f32(false, a, false, b,
                                               (short)0, c, false, false);
}

// A operand: A[row=m0+lane%16][k]; weight stored [k][m] row-major, stride ldm.
__device__ __forceinline__ v2f frag_a(const float* __restrict__ w, int ldm,
                                      int m0, int k, int lane) {
  const int row = m0 + (lane & 15);
  const int kk  = k + ((lane >> 4) << 1);
  v2f a;
  a.x = w[kk * ldm + row];
  a.y = w[(kk + 1) * ldm + row];
  return a;
}

// B operand: B[k][col=n0+lane%16]; activations stored [k][n], stride ldn.
__device__ __forceinline__ v2f frag_b(const float* __restrict__ act, int ldn,
                                      int n0, int k, int lane) {
  const int col = n0 + (lane & 15);
  const int kk  = k + (lane >> 4);
  v2f b;
  b.x = act[kk * ldn + col];
  b.y = act[(kk + 2) * ldn + col];
  return b;
}

// --- Tensor Data Mover: DMA a (tile_h x tile_w) f32 tile (row stride
// row_stride elements) from global memory into LDS, packed contiguously.
// D# bit layout per CDNA5 ISA ch. 10.8 (group0: count/lds/global/type,
// group1: data_size, tensor dims, tile dims, dim0 stride).
__device__ __forceinline__ void tdm_load_tile(const float* __restrict__ gsrc,
                                              unsigned lds_off,
                                              unsigned tile_w, unsigned tile_h,
                                              unsigned row_stride) {
  const unsigned long long ga = (unsigned long long)(size_t)gsrc;
  u32x4 g0;
  g0[0] = 1u;                                              // count=1, user desc
  g0[1] = lds_off;                                         // lds_addr (bytes)
  g0[2] = (unsigned)ga;                                    // global_addr[31:0]
  g0[3] = (unsigned)((ga >> 32) & 0x01FFFFFFu) | (2u << 30); // addr[56:32]|type=2
  i32x8 g1;
  g1[0] = 0x20000;                                         // data_size=4B
  g1[1] = (int)((tile_w & 0xFFFFu) << 16);                 // tensor_dim0 lo16
  g1[2] = (int)(((tile_w >> 16) & 0xFFFFu) |               // tensor_dim0 hi16
                ((tile_h & 0xFFFFu) << 16));               // tensor_dim1 lo16
  g1[3] = (int)(((tile_h >> 16) & 0xFFFFu) |               // tensor_dim1 hi16
                ((tile_w & 0xFFFFu) << 16));               // tile_dim0
  g1[4] = (int)(tile_h & 0xFFFFu);                         // tile_dim1 (dim2=0)
  g1[5] = (int)row_stride;                                 // dim0_stride[31:0]
  g1[6] = 0;
  g1[7] = 0;
  i32x4 g2 = {0, 0, 0, 0};
  i32x4 g3 = {0, 0, 0, 0};
#if __clang_major__ >= 23
  i32x8 g4 = {0, 0, 0, 0, 0, 0, 0, 0};
  __builtin_amdgcn_tensor_load_to_lds(g0, g1, g2, g3, g4, 0);
#else
  __builtin_amdgcn_tensor_load_to_lds(g0, g1, g2, g3, 0);
#endif
}

// One 128 x NCV GEMM stage: out = W^T * act (optionally gated per channel).
template <bool GATED>
__device__ __forceinline__ void vec_stage(const float* __restrict__ wlds,
                                          const float* __restrict__ actb,
                                          float* __restrict__ outb,
                                          const float* __restrict__ gate,
                                          int wave, int lane) {
  v8f acc0 = {}, acc1 = {}, acc2 = {};
  const int m0 = wave * 16;
  #pragma unroll 4
  for (int k = 0; k < S_DIM; k += 4) {
    v2f a  = frag_a(wlds, S_DIM, m0, k, lane);   // A reused across 3 N-tiles
    v2f b0 = frag_b(actb, NCV, 0,  k, lane);
    v2f b1 = frag_b(actb, NCV, 16, k, lane);
    v2f b2 = frag_b(actb, NCV, 32, k, lane);
    acc0 = wmma_f32_k4(a, b0, acc0);
    acc1 = wmma_f32_k4(a, b1, acc1);
    acc2 = wmma_f32_k4(a, b2, acc2);
  }
  const int col0  = lane & 15;
  const int rbase = m0 + ((lane >> 4) << 3);
  #pragma unroll
  for (int i = 0; i < 8; ++i) {
    const int r = rbase + i;
    float g0 = 1.f, g1 = 1.f, g2 = 1.f;
    if (GATED) {
      g0 = gate[r * TOK + ((0  + col0) / 3)];
      g1 = gate[r * TOK + ((16 + col0) / 3)];
      g2 = gate[r * TOK + ((32 + col0) / 3)];
    }
    outb[r * NCV + 0  + col0] = acc0[i] * g0;
    outb[r * NCV + 16 + col0] = acc1[i] * g1;
    outb[r * NCV + 32 + col0] = acc2[i] * g2;
  }
}

// Final projection half: out cols n = half*128 + [0,128), weight half in LDS
// as [k=0..127][n_local=0..127].
__device__ __forceinline__ void final_stage(const float* __restrict__ wlds,
                                            const float* __restrict__ vb,
                                            float* __restrict__ out,
                                            long tok0, int half,
                                            int wave, int lane) {
  const float inv_sqrt = 0.0883883476483184f;   // 1/sqrt(128)
  const int m0 = wave * 16;
  v8f a0 = {}, a1 = {}, a2 = {};
  #pragma unroll 4
  for (int k = 0; k < S_DIM; k += 4) {
    v2f a  = frag_a(wlds, S_DIM, m0, k, lane);
    v2f b0 = frag_b(vb, NCV, 0,  k, lane);
    v2f b1 = frag_b(vb, NCV, 16, k, lane);
    v2f b2 = frag_b(vb, NCV, 32, k, lane);
    a0 = wmma_f32_k4(a, b0, a0);
    a1 = wmma_f32_k4(a, b1, a1);
    a2 = wmma_f32_k4(a, b2, a2);
  }
  const int col0  = lane & 15;
  const int rbase = half * 128 + m0 + ((lane >> 4) << 3);
  #pragma unroll
  for (int i = 0; i < 8; ++i) {
    const int n = rbase + i;
    const int g = n >> 4, j = n & 15;
    {
      const int cn = 0 + col0, t = cn % 3, tk = cn / 3;
      out[(tok0 + tk) * 1024 + g * 64 + 16 + j * 3 + t] = a0[i] * inv_sqrt;
    }
    {
      const int cn = 16 + col0, t = cn % 3, tk = cn / 3;
      out[(tok0 + tk) * 1024 + g * 64 + 16 + j * 3 + t] = a1[i] * inv_sqrt;
    }
    {
      const int cn = 32 + col0, t = cn % 3, tk = cn / 3;
      out[(tok0 + tk) * 1024 + g * 64 + 16 + j * 3 + t] = a2[i] * inv_sqrt;
    }
  }
}

__global__ void __launch_bounds__(THREADS)
gvp_backflow_kernel(const float* __restrict__ mo,
                    const float* __restrict__ Wh0,
                    const float* __restrict__ Wu0,
                    const float* __restrict__ Wm0,
                    const float* __restrict__ bm0,
                    const float* __restrict__ Wh1,
                    const float* __restrict__ Wu1,
                    const float* __restrict__ W1,
                    float* __restrict__ out) {
  extern __shared__ float lds[];
  float* wbufA = lds;                          // 16384 f (64 KB) weight buf A
  float* wbufB = wbufA + S_DIM * S_DIM;        // 16384 f (64 KB) weight buf B
  float* vbuf  = wbufB + S_DIM * S_DIM;        // [128][48]
  float* hbuf  = vbuf + S_DIM * NCV;           // [128][48]
  float* smat  = hbuf + S_DIM * NCV;           // [256][16]  ([s ; vn])
  float* gate  = smat + 2 * S_DIM * TOK;       // [128][16]

  const int  tid  = threadIdx.x;
  const int  lane = tid & 31;
  const int  wv   = tid >> 5;
  const long tok0 = (long)blockIdx.x * TOK;
  const unsigned ldsA = (unsigned)(size_t)wbufA;
  const unsigned ldsB = (unsigned)(size_t)wbufB;

  // Kick off the first two weight DMAs before touching anything else.
  if (wv == 0) {
    tdm_load_tile(Wh0, ldsA, S_DIM * S_DIM, 1, S_DIM * S_DIM);        // J0 -> A
    tdm_load_tile(Wm0, ldsB, S_DIM * S_DIM, 1, S_DIM * S_DIM);        // J1 -> B
  }

  // Scalar output slots are exactly zero (antisymmetry); overlap with DMA.
  for (int i = tid; i < TOK * 256; i += THREADS) {
    const int t = i >> 8, r = i & 255;
    out[(tok0 + t) * 1024 + ((r >> 4) << 6) + (r & 15)] = 0.f;
  }

  // Stage activations: s -> smat[0:128][tok], v -> vbuf[m][tok*3+t].
  for (int i = tid; i < TOK * S_DIM; i += THREADS) {
    const int t = i >> 7, c = i & 127;
    smat[c * TOK + t] = mo[(tok0 + t) * 512 + c];
  }
  for (int i = tid; i < TOK * S_DIM * 3; i += THREADS) {
    const int t = i / (S_DIM * 3);
    const int r = i - t * (S_DIM * 3);                     // r = m*3 + xyz
    vbuf[(r / 3) * NCV + t * 3 + (r % 3)] = mo[(tok0 + t) * 512 + S_DIM + r];
  }

  // ---- stage 0 (A): Vh0 = Wh0^T v ----
  if (wv == 0) __builtin_amdgcn_s_wait_tensorcnt(1);
  __syncthreads();
  vec_stage<false>(wbufA, vbuf, hbuf, nullptr, wv, lane);
  __syncthreads();
  if (wv == 0)                                             // J2 -> A (Wm0 hi)
    tdm_load_tile(Wm0 + S_DIM * S_DIM, ldsA, S_DIM * S_DIM, 1, S_DIM * S_DIM);

  // vn into smat[128:256][tok] (overlaps J2 DMA).
  for (int i = tid; i < S_DIM * TOK; i += THREADS) {
    const int h = i / TOK, t = i - (i / TOK) * TOK;
    const float x = hbuf[h * NCV + t * 3 + 0];
    const float y = hbuf[h * NCV + t * 3 + 1];
    const float z = hbuf[h * NCV + t * 3 + 2];
    smat[(S_DIM + h) * TOK + t] = sqrtf(x * x + y * y + z * z + 1e-8f);
  }

  // ---- stage 1 (B): gate pass 1, K = s rows 0..127 ----
  if (wv == 0) __builtin_amdgcn_s_wait_tensorcnt(1);
  __syncthreads();
  v8f gacc = {};
  {
    const int m0 = wv * 16;
    #pragma unroll 4
    for (int k = 0; k < S_DIM; k += 4) {
      v2f a = frag_a(wbufB, S_DIM, m0, k, lane);
      v2f b = frag_b(smat, TOK, 0, k, lane);
      gacc = wmma_f32_k4(a, b, gacc);
    }
  }
  __syncthreads();
  if (wv == 0) tdm_load_tile(Wu0, ldsB, S_DIM * S_DIM, 1, S_DIM * S_DIM); // J3

  // ---- stage 2 (A): gate pass 2, K = vn rows 128..255, then sigmoid ----
  if (wv == 0) __builtin_amdgcn_s_wait_tensorcnt(1);
  __syncthreads();
  {
    const int m0 = wv * 16;
    #pragma unroll 4
    for (int k = 0; k < S_DIM; k += 4) {
      v2f a = frag_a(wbufA, S_DIM, m0, k, lane);
      v2f b = frag_b(smat + S_DIM * TOK, TOK, 0, k, lane);
      gacc = wmma_f32_k4(a, b, gacc);
    }
    const int col   = lane & 15;
    const int rbase = m0 + ((lane >> 4) << 3);
    #pragma unroll
    for (int i = 0; i < 8; ++i) {
      const float sm = gacc[i] + bm0[rbase + i];
      gate[(rbase + i) * TOK + col] = 1.f / (1.f + __expf(-sm));
    }
  }
  __syncthreads();
  if (wv == 0) tdm_load_tile(Wh1, ldsA, S_DIM * S_DIM, 1, S_DIM * S_DIM); // J4

  // ---- stage 3 (B): v' = (Wu0^T Vh0) * gate ----
  if (wv == 0) __builtin_amdgcn_s_wait_tensorcnt(1);
  __syncthreads();
  vec_stage<true>(wbufB, hbuf, vbuf, gate, wv, lane);
  __syncthreads();
  if (wv == 0) tdm_load_tile(Wu1, ldsB, S_DIM * S_DIM, 1, S_DIM * S_DIM); // J5

  // ---- stage 4 (A): Vh1 = Wh1^T v' ----
  if (wv == 0) __builtin_amdgcn_s_wait_tensorcnt(1);
  __syncthreads();
  vec_stage<false>(wbufA, vbuf, hbuf, nullptr, wv, lane);
  __syncthreads();
  if (wv == 0) tdm_load_tile(W1, ldsA, 128, 128, MOUT);    // J6: W1 cols 0..127

  // ---- stage 5 (B): Vu1 = Wu1^T Vh1 ----
  if (wv == 0) __builtin_amdgcn_s_wait_tensorcnt(1);
  __syncthreads();
  vec_stage<false>(wbufB, hbuf, vbuf, nullptr, wv, lane);
  __syncthreads();
  if (wv == 0) tdm_load_tile(W1 + 128, ldsB, 128, 128, MOUT); // J7: cols 128..255

  // ---- stage 6 (A): vo cols 0..127 -> global ----
  if (wv == 0) __builtin_amdgcn_s_wait_tensorcnt(1);
  __syncthreads();
  final_stage(wbufA, vbuf, out, tok0, 0, wv, lane);

  // ---- stage 7 (B): vo cols 128..255 -> global ----
  if (wv == 0) __builtin_amdgcn_s_wait_tensorcnt(0);
  __syncthreads();
  final_stage(wbufB, vbuf, out, tok0, 1, wv, lane);
}

extern "C" void kernel_launch(void* const* d_in, const int* in_sizes, int n_in,
                              void* d_out, int out_size, void* d_ws, size_t ws_size,
                              hipStream_t stream) {
  const float* mo  = (const float*)d_in[0];
  const float* Wh0 = (const float*)d_in[1];
  const float* Wu0 = (const float*)d_in[2];
  const float* Wm0 = (const float*)d_in[3];
  const float* bm0 = (const float*)d_in[4];
  const float* Wh1 = (const float*)d_in[5];
  const float* Wu1 = (const float*)d_in[6];
  // d_in[7]=Wm1, d_in[8]=bm1, d_in[9]=W0: provably dead (scalar branch == 0).
  const float* W1  = (const float*)d_in[10];
  float* out = (float*)d_out;

  const int ntok = in_sizes[0] / 512;      // 32768
  const int grid = ntok / TOK;             // 2048

  const size_t shmem =
      (size_t)(2 * S_DIM * S_DIM + 2 * S_DIM * NCV + 2 * S_DIM * TOK +
               S_DIM * TOK) * sizeof(float);   // 204800 B, < 320 KB WGP LDS

  hipFuncSetAttribute(reinterpret_cast<const void*>(gvp_backflow_kernel),
                      hipFuncAttributeMaxDynamicSharedMemorySize, (int)shmem);

  gvp_backflow_kernel<<<grid, THREADS, shmem, stream>>>(
      mo, Wh0, Wu0, Wm0, bm0, Wh1, Wu1, W1, out);
}